// TemporalEncoderDecoderAttention_3839700763043
// MI455X (gfx1250) — compile-verified
//
#include <hip/hip_runtime.h>

// ---- problem constants (from reference) ----
#define BB 4
#define P1 12
#define PP 12
#define NN 207
#define DM 128     // d_model
#define DK 16      // d_k
#define HH 8       // heads
#define DH 256     // d_hid
#define TOKENS (BB*P1*NN)          // 9936
#define MT 32                      // tokens per workgroup (2 WMMA M-tiles)
#define NWG ((TOKENS + MT - 1)/MT) // 311 (last WG handles 16 valid tokens)

typedef float v2f __attribute__((ext_vector_type(2)));
typedef float v8f __attribute__((ext_vector_type(8)));

// D = A(16x4 f32) * B(4x16 f32) + C(16x16 f32), full fp32 WMMA
__device__ __forceinline__ v8f wmma_k4(v2f a, v2f b, v8f c) {
  return __builtin_amdgcn_wmma_f32_16x16x4_f32(
      /*neg_a=*/false, a, /*neg_b=*/false, b,
      /*c_mod=*/(short)0, c, /*reuse_a=*/false, /*reuse_b=*/false);
}

__global__ __launch_bounds__(256, 1)
void ted_attn_fused(const float* __restrict__ inputs,     // (B,P1,N,DM)
                    const float* __restrict__ enc_K,      // (B,P,N,H,DK)
                    const float* __restrict__ enc_V,      // (B,P,N,H,DK)
                    const float* __restrict__ c_targets,  // (B,P1,N,DM)
                    const float* __restrict__ W1,         // (DH,DM)
                    const float* __restrict__ b1,         // (DH)
                    const float* __restrict__ W2,         // (H*DK*DM, DH)
                    const float* __restrict__ b2,         // (H*DK*DM)
                    const float* __restrict__ W_out,      // (DM,DM)
                    const float* __restrict__ gamma,
                    const float* __restrict__ beta,
                    float* __restrict__ out)              // (B,P1,N,DM)
{
  const int wg   = blockIdx.x;          // 0..310
  const int tid  = threadIdx.x;         // 0..255
  const int wave = tid >> 5;            // 0..7
  const int lane = tid & 31;
  const int hi   = lane >> 4;           // lane half (0/1)
  const int ln   = lane & 15;           // lane-in-half

  // 64 KB LDS, 3 regions with barrier-separated lifetimes:
  //  R0 (16KB): c_targets tile (32x128)  -> later attention output (32x128)
  //  R1 (32KB): hidden H (32x256)
  //  R2 (16KB): Q accumulator (32x128)   -> later proj+residual (32x128)
  __shared__ float smem[16384];
  float* R0 = smem;
  float* R1 = smem + 4096;
  float* R2 = smem + 12288;

  const int tok0 = wg * MT;

  // ---- initial: coalesced c-tile load (clamped) + zero Q ----
  {
    const float4* cin = (const float4*)c_targets;
    float4* r0 = (float4*)R0;
    const int lim = TOKENS * (DM/4) - 1;
    for (int i = tid; i < MT*DM/4; i += 256) {
      int g = tok0 * (DM/4) + i;
      r0[i] = cin[g > lim ? lim : g];
    }
    for (int i = tid; i < MT*DM; i += 256) R2[i] = 0.0f;
  }
  __syncthreads();

  // ============ Stage A: H = relu(C @ W1^T + b1), (32 x 256) ============
  // M=32 (2 tiles share each B tile), N: wave owns 2x16 hid cols, K=128.
  for (int nt = 0; nt < 2; ++nt) {
    const int nb = wave * 32 + nt * 16;
    v8f acc0 = {}, acc1 = {};
    const float* brow = W1 + (size_t)(nb + ln) * DM;
    const float* ar0  = R0 + ln * DM;
    const float* ar1  = R0 + (16 + ln) * DM;
    for (int kb = 0; kb < DM; kb += 4) {
      const int ka = kb + 2 * hi;
      v2f b;  { float2 t = *(const float2*)(brow + ka); b.x  = t.x; b.y  = t.y; }
      v2f a0; { float2 t = *(const float2*)(ar0  + ka); a0.x = t.x; a0.y = t.y; }
      v2f a1; { float2 t = *(const float2*)(ar1  + ka); a1.x = t.x; a1.y = t.y; }
      acc0 = wmma_k4(a0, b, acc0);
      acc1 = wmma_k4(a1, b, acc1);
    }
    const float b1v = b1[nb + ln];
    #pragma unroll
    for (int v = 0; v < 8; ++v) {
      R1[(v + 8*hi) * DH + nb + ln]        = fmaxf(acc0[v] + b1v, 0.0f);
      R1[(16 + v + 8*hi) * DH + nb + ln]   = fmaxf(acc1[v] + b1v, 0.0f);
    }
  }
  __syncthreads();

  // preload x values this lane needs in the j-loop (clamped, constant over j)
  const int db = wave * 16;                   // d-range of this wave
  float xreg[16];
  #pragma unroll
  for (int mt = 0; mt < 2; ++mt)
    #pragma unroll
    for (int v = 0; v < 8; ++v) {
      int T = tok0 + mt*16 + v + 8*hi;
      if (T > TOKENS - 1) T = TOKENS - 1;
      xreg[mt*8 + v] = inputs[(size_t)T * DM + db + ln];
    }

  // ====== Stage B (hot): Q[t][j] = sum_d (H@W2_j^T + b2)[t][d] * x[t][d] ======
  // Per j: 2 M-tiles share each W2 B-tile -> 128 WMMAs per 64 global b64 loads.
  for (int j = 0; j < HH * DK; ++j) {
    v8f acc0 = {}, acc1 = {};
    const float* w2row = W2 + (size_t)(j * DM + db + ln) * DH;
    const float* ar0   = R1 + ln * DH;
    const float* ar1   = R1 + (16 + ln) * DH;
    for (int kb = 0; kb < DH; kb += 4) {
      const int ka = kb + 2 * hi;
      v2f b;  { float2 t = *(const float2*)(w2row + ka); b.x  = t.x; b.y  = t.y; }
      v2f a0; { float2 t = *(const float2*)(ar0   + ka); a0.x = t.x; a0.y = t.y; }
      v2f a1; { float2 t = *(const float2*)(ar1   + ka); a1.x = t.x; a1.y = t.y; }
      acc0 = wmma_k4(a0, b, acc0);
      acc1 = wmma_k4(a1, b, acc1);
    }
    const float b2v = b2[j * DM + db + ln];
    float s[16];
    #pragma unroll
    for (int v = 0; v < 8; ++v) {
      s[v]     = (acc0[v] + b2v) * xreg[v];
      s[8 + v] = (acc1[v] + b2v) * xreg[8 + v];
    }
    #pragma unroll
    for (int off = 1; off < 16; off <<= 1) {
      #pragma unroll
      for (int v = 0; v < 16; ++v) s[v] += __shfl_xor(s[v], off, 16);
    }
    if (ln == 0) {
      #pragma unroll
      for (int v = 0; v < 8; ++v) {
        atomicAdd(&R2[(v + 8*hi) * DM + j],      s[v]);
        atomicAdd(&R2[(16 + v + 8*hi) * DM + j], s[8 + v]);
      }
    }
  }
  __syncthreads();

  // ============ Stage C: attention over P=12 (tiny; 32 tok x 8 heads) ============
  {
    const int t = tid >> 3;                 // 0..31
    const int h = tid & 7;
    int T = tok0 + t; if (T > TOKENS - 1) T = TOKENS - 1;
    const int b = T / (P1 * NN);
    const int n = (T % (P1 * NN)) % NN;

    float qv[DK];
    #pragma unroll
    for (int k = 0; k < DK; ++k) qv[k] = R2[t * DM + h * DK + k];

    float sc[PP];
    float mx = -1e30f;
    for (int p = 0; p < PP; ++p) {
      const float* kp = enc_K + ((((size_t)b * PP + p) * NN + n) * HH + h) * DK;
      float d = 0.0f;
      #pragma unroll
      for (int k = 0; k < DK; ++k) d += qv[k] * kp[k];
      sc[p] = d * 0.25f;                    // 1/sqrt(16)
      mx = fmaxf(mx, sc[p]);
    }
    float sum = 0.0f;
    for (int p = 0; p < PP; ++p) { sc[p] = __expf(sc[p] - mx); sum += sc[p]; }
    const float inv = 1.0f / sum;

    float o[DK];
    #pragma unroll
    for (int k = 0; k < DK; ++k) o[k] = 0.0f;
    for (int p = 0; p < PP; ++p) {
      const float* vp = enc_V + ((((size_t)b * PP + p) * NN + n) * HH + h) * DK;
      const float a = sc[p] * inv;
      #pragma unroll
      for (int k = 0; k < DK; ++k) o[k] += a * vp[k];
    }
    #pragma unroll
    for (int k = 0; k < DK; ++k) R0[t * DM + h * DK + k] = o[k];  // R0 reused
  }
  __syncthreads();

  // ============ Stage D: out = attn @ W_out^T + x (WMMA, 2 M-tiles) ============
  {
    const int nb = wave * 16;
    v8f acc0 = {}, acc1 = {};
    const float* brow = W_out + (size_t)(nb + ln) * DM;
    const float* ar0  = R0 + ln * DM;
    const float* ar1  = R0 + (16 + ln) * DM;
    for (int kb = 0; kb < DM; kb += 4) {
      const int ka = kb + 2 * hi;
      v2f b;  { float2 t = *(const float2*)(brow + ka); b.x  = t.x; b.y  = t.y; }
      v2f a0; { float2 t = *(const float2*)(ar0  + ka); a0.x = t.x; a0.y = t.y; }
      v2f a1; { float2 t = *(const float2*)(ar1  + ka); a1.x = t.x; a1.y = t.y; }
      acc0 = wmma_k4(a0, b, acc0);
      acc1 = wmma_k4(a1, b, acc1);
    }
    #pragma unroll
    for (int v = 0; v < 8; ++v) {
      const int m0 = v + 8*hi, m1 = 16 + v + 8*hi;
      int T0 = tok0 + m0; if (T0 > TOKENS - 1) T0 = TOKENS - 1;
      int T1 = tok0 + m1; if (T1 > TOKENS - 1) T1 = TOKENS - 1;
      R2[m0 * DM + nb + ln] = acc0[v] + inputs[(size_t)T0 * DM + nb + ln];
      R2[m1 * DM + nb + ln] = acc1[v] + inputs[(size_t)T1 * DM + nb + ln];
    }
  }
  __syncthreads();

  // ============ Stage E: LayerNorm + guarded store ============
  #pragma unroll
  for (int half = 0; half < 2; ++half) {
    const int t = 4*wave + 2*half + hi;     // 0..31
    float vals[8], s1 = 0.0f, s2 = 0.0f;
    #pragma unroll
    for (int e = 0; e < 8; ++e) {
      const float v = R2[t * DM + ln + 16*e];
      vals[e] = v; s1 += v; s2 += v * v;
    }
    #pragma unroll
    for (int off = 1; off < 16; off <<= 1) {
      s1 += __shfl_xor(s1, off, 16);
      s2 += __shfl_xor(s2, off, 16);
    }
    const float mu   = s1 * (1.0f / DM);
    const float var  = s2 * (1.0f / DM) - mu * mu;
    const float rstd = rsqrtf(var + 1e-5f);
    const int T = tok0 + t;
    if (T < TOKENS) {
      float* op = out + (size_t)T * DM;
      #pragma unroll
      for (int e = 0; e < 8; ++e) {
        const int d = ln + 16*e;
        op[d] = (vals[e] - mu) * rstd * gamma[d] + beta[d];
      }
    }
  }
}

extern "C" void kernel_launch(void* const* d_in, const int* in_sizes, int n_in,
                              void* d_out, int out_size, void* d_ws, size_t ws_size,
                              hipStream_t stream) {
  (void)in_sizes; (void)n_in; (void)out_size; (void)d_ws; (void)ws_size;
  const float* inputs    = (const float*)d_in[0];
  const float* enc_K     = (const float*)d_in[1];
  const float* enc_V     = (const float*)d_in[2];
  const float* c_targets = (const float*)d_in[3];
  const float* W1        = (const float*)d_in[4];
  const float* b1        = (const float*)d_in[5];
  const float* W2        = (const float*)d_in[6];
  const float* b2        = (const float*)d_in[7];
  const float* W_out     = (const float*)d_in[8];
  const float* gamma     = (const float*)d_in[9];
  const float* beta      = (const float*)d_in[10];
  float* outp            = (float*)d_out;

  hipLaunchKernelGGL(ted_attn_fused, dim3(NWG), dim3(256), 0, stream,
                     inputs, enc_K, enc_V, c_targets,
                     W1, b1, W2, b2, W_out, gamma, beta, outp);
}